// SP_Loss_20787641712926
// MI455X (gfx1250) — compile-verified
//
#include <hip/hip_runtime.h>
#include <stdint.h>

// --------------------------------------------------------------------------
// SP loss on MI455X (gfx1250):
//  - HBM -> LDS staging via Tensor Data Mover (tensor_load_to_lds), with the
//    D# padding feature producing a bank-conflict-free stride-132 layout,
//    double-buffered across K-chunks (TENSORcnt overlap).
//  - In-place LDS sqrt pass (one v_sqrt_f32 per element).
//  - Per-batch Gram matrix with V_WMMA_F32_16X16X4_F32 (full fp32).
//  - Deterministic two-stage reduction (no float atomics).
// Shapes: sp_v [B=4096, A=64, D=512] f32 -> scalar f32.
// Roofline: 537 MB / 23.3 TB/s ~ 23 us floor; 17.2 GFLOP -> 0.75 TFLOPS
// needed -> purely bandwidth-bound, so fp32 WMMA + TDM is the right path.
// --------------------------------------------------------------------------

typedef float v2f __attribute__((ext_vector_type(2)));
typedef float v8f __attribute__((ext_vector_type(8)));
typedef unsigned int u32x4 __attribute__((ext_vector_type(4)));
typedef int i32x4 __attribute__((ext_vector_type(4)));
typedef int i32x8 __attribute__((ext_vector_type(8)));

#define A_DIM      64
#define D_DIM      512
#define CHUNK      128
#define NCHUNK     (D_DIM / CHUNK)
#define LDS_STRIDE (CHUNK + 4)      // 132 floats: 16B-aligned rows, 132%64=4
                                    // -> fragment banks 4*row+k all distinct
#define THREADS    512
#define WAVES      (THREADS / 32)   // 16 wave32 waves -> 16 G-tiles of 16x16

// TDM load of one 64x128 f32 tile (row stride 512 elems in memory) into LDS
// with 16B padding after every 512B row -> stride-132-float rows in LDS.
// D# layout per CDNA5 ISA section 8 (groups 0..3).
__device__ __forceinline__ void tdm_load_chunk(const float* gsrc, void* ldsDst)
{
    const uint64_t ga = (uint64_t)(uintptr_t)gsrc;
    const uint32_t la = (uint32_t)(uintptr_t)ldsDst;   // low 32b = LDS byte offset

    u32x4 g0;
    g0[0] = 1u;                                        // count=1, user descriptor
    g0[1] = la;                                        // lds_addr
    g0[2] = (uint32_t)ga;                              // global_addr[31:0]
    g0[3] = (uint32_t)((ga >> 32) & 0x01FFFFFFu)       // global_addr[56:32]
          | 0x80000000u;                               // type=2 ("image")

    i32x8 g1;
    g1[0] = (2 << 16)      // data_size = 4 bytes
          | (1 << 20)      // pad_enable
          | (6 << 22)      // pad_interval: 2^(6+1) = 128 DWORDs (= one row)
          | (3 << 25);     // pad_amount: 3+1 = 4 DWORDs (16 bytes)
                           // workgroup_mask = 0 (not in a cluster)
    g1[1] = (int)(512u << 16);   // tensor_dim0[15:0] = 512 (row length)
    g1[2] = (int)(64u  << 16);   // tensor_dim1[15:0] = 64  (rows)
    g1[3] = (int)(128u << 16);   // tile_dim0 = 128 columns
    g1[4] = 64;                  // tile_dim1 = 64 rows, tile_dim2 = 0
    g1[5] = 512;                 // tensor_dim0_stride[31:0] = 512 elems
    g1[6] = 0;
    g1[7] = 0;

    i32x4 g2 = {0, 0, 0, 0};     // 2-D tensor: groups 2/3 unused
    i32x4 g3 = {0, 0, 0, 0};
    i32x8 g4 = {0, 0, 0, 0, 0, 0, 0, 0};   // trailing group (unused here)

    __builtin_amdgcn_tensor_load_to_lds(g0, g1, g2, g3, g4, /*cpol=*/0);
}

__global__ __launch_bounds__(THREADS)
void sp_loss_batch_kernel(const float* __restrict__ spv,
                          float* __restrict__ partial)
{
    __shared__ float sS[2][A_DIM * LDS_STRIDE];  // double-buffered 64x132 f32
    __shared__ float sqn[A_DIM];                 // row norms = diag(G)
    __shared__ float wsum[WAVES];

    const int b    = blockIdx.x;
    const int tid  = threadIdx.x;
    const int lane = tid & 31;
    const int wave = tid >> 5;
    const int ti   = wave >> 2;                  // G tile row (0..3)
    const int tj   = wave & 3;                   // G tile col (0..3)

    const float* base = spv + (size_t)b * (A_DIM * D_DIM);

    // WMMA f32 16x16x4 operand addressing (ISA 7.12.2, 32-bit A 16x4):
    // lane l<16 : M=l, K=k..k+1 ; lane l>=16 : M=l-16, K=k+2..k+3.
    // B = S^T of the same matrix -> identical per-lane gather from tj block.
    const int colShift = (lane >> 4) << 1;       // 0 or 2
    const int rA = ti * 16 + (lane & 15);
    const int rB = tj * 16 + (lane & 15);

    // ---- prime the pipeline: TDM chunk 0 -> buffer 0 ----
    if (wave == 0) {
        tdm_load_chunk(base, &sS[0][0]);
        __builtin_amdgcn_s_wait_tensorcnt(0);
    }
    __syncthreads();

    v8f c = {};                                  // 16x16 f32 accumulator tile

    for (int ch = 0; ch < NCHUNK; ++ch) {
        float* buf = &sS[ch & 1][0];

        // ---- kick off DMA for next chunk into the other buffer ----
        if ((ch + 1) < NCHUNK && wave == 0)
            tdm_load_chunk(base + (ch + 1) * CHUNK, &sS[(ch + 1) & 1][0]);

        // ---- in-place sqrt pass: raw sp_v -> S (float4 per thread x4) ----
        #pragma unroll
        for (int it = 0; it < (A_DIM * CHUNK) / (THREADS * 4); ++it) {
            const int idx4 = it * THREADS + tid;         // 0..2047
            const int r    = idx4 >> 5;                  // /32 float4 per row
            const int c4   = idx4 & 31;
            float4* p = (float4*)&buf[r * LDS_STRIDE + c4 * 4];
            float4 v = *p;
            v.x = sqrtf(v.x); v.y = sqrtf(v.y);
            v.z = sqrtf(v.z); v.w = sqrtf(v.w);
            *p = v;
        }
        __syncthreads();

        // ---- K-loop: 32 x V_WMMA_F32_16X16X4_F32 per chunk ----
        const float* Ap = &buf[rA * LDS_STRIDE + colShift];
        const float* Bp = &buf[rB * LDS_STRIDE + colShift];
        #pragma unroll 8
        for (int k = 0; k < CHUNK; k += 4) {
            v2f a, bb;
            a.x  = Ap[k];     a.y  = Ap[k + 1];
            bb.x = Bp[k];     bb.y = Bp[k + 1];
            c = __builtin_amdgcn_wmma_f32_16x16x4_f32(
                    /*neg_a=*/false, a, /*neg_b=*/false, bb,
                    /*c_mod=*/(short)0, c,
                    /*reuse_a=*/false, /*reuse_b=*/false);
        }

        // ---- publish next chunk's DMA completion, recycle buffers ----
        if ((ch + 1) < NCHUNK && wave == 0)
            __builtin_amdgcn_s_wait_tensorcnt(0);
        __syncthreads();
    }

    // ---- row norms from diag(G): VGPR v holds M=v (lanes 0-15) /
    // M=v+8 (lanes 16-31), N = lane%16 (ISA 7.12.2). ----
    if (ti == tj) {
        #pragma unroll
        for (int r = 0; r < 8; ++r) {
            const float v0 = __shfl(c[r], r,      32);   // (M=r,   N=r)
            const float v1 = __shfl(c[r], r + 24, 32);   // (M=r+8, N=r+8)
            if (lane == 0) {
                sqn[ti * 16 + r]     = v0;
                sqn[ti * 16 + 8 + r] = v1;
            }
        }
    }
    __syncthreads();

    // ---- pairwise distances over strict upper triangle ----
    float dsum = 0.0f;
    const int jcol  = tj * 16 + (lane & 15);
    const int ibase = ti * 16 + ((lane >> 4) << 3);
    const float sqj = sqn[jcol];
    #pragma unroll
    for (int r = 0; r < 8; ++r) {
        const int i = ibase + r;
        if (i < jcol) {
            const float d2 = sqn[i] + sqj - 2.0f * c[r];
            dsum += sqrtf(fmaxf(d2, 0.0f));
        }
    }

    // wave32 butterfly reduce
    #pragma unroll
    for (int off = 16; off > 0; off >>= 1)
        dsum += __shfl_xor(dsum, off, 32);
    if (lane == 0) wsum[wave] = dsum;
    __syncthreads();

    // deterministic per-block finish (fixed sequential order)
    if (tid == 0) {
        float tot = 0.0f;
        #pragma unroll
        for (int w = 0; w < WAVES; ++w) tot += wsum[w];
        const float scale = 1.0f / (1.41421356237309515f * 2016.0f); // 1/(sqrt2*cnt)
        partial[b] = 1.0f - tot * scale;
    }
}

// Deterministic fixed-tree reduction of per-batch values -> scalar.
__global__ __launch_bounds__(1024)
void sp_loss_reduce_kernel(const float* __restrict__ partial,
                           float* __restrict__ out, int n)
{
    __shared__ float sm[1024];
    const int tid = threadIdx.x;
    float s = 0.0f;
    for (int i = tid; i < n; i += 1024) s += partial[i];
    sm[tid] = s;
    __syncthreads();
    #pragma unroll
    for (int step = 512; step > 0; step >>= 1) {
        if (tid < step) sm[tid] += sm[tid + step];
        __syncthreads();
    }
    if (tid == 0) out[0] = sm[0];
}

extern "C" void kernel_launch(void* const* d_in, const int* in_sizes, int n_in,
                              void* d_out, int out_size, void* d_ws, size_t ws_size,
                              hipStream_t stream)
{
    const float* spv = (const float*)d_in[0];
    float* out       = (float*)d_out;
    float* partial   = (float*)d_ws;                  // B floats of scratch

    const int B = in_sizes[0] / (A_DIM * D_DIM);      // 4096

    sp_loss_batch_kernel<<<B, THREADS, 0, stream>>>(spv, partial);
    sp_loss_reduce_kernel<<<1, 1024, 0, stream>>>(partial, out, B);
}